// EmbeddingLoss_37280316129706
// MI455X (gfx1250) — compile-verified
//
#include <hip/hip_runtime.h>
#include <hip/hip_bf16.h>
#include <math.h>

#define NB    8
#define ND    64
#define NH_   256
#define NW    256
#define NS    (NH_ * NW)       // 65536, power of two
#define NK    17
#define NMAX  8
#define NPAD  16               // WMMA M/N padding
#define SIG2INV 100.0f         // 1/(0.1*0.1)

typedef __attribute__((ext_vector_type(2))) float v2f;
typedef __attribute__((ext_vector_type(8))) float v8f;

// ---------------------------------------------------------------------------
// Kernel 0: zero the e-workspace [B,N,K,D] and the scalar output.
// ---------------------------------------------------------------------------
__global__ void ae_zero(float* __restrict__ e, float* __restrict__ out) {
  int i = blockIdx.x * blockDim.x + threadIdx.x;
  if (i < NB * NMAX * NK * ND) e[i] = 0.0f;
  if (i == 0) out[0] = 0.0f;
}

// ---------------------------------------------------------------------------
// Kernel 1: scan tags (int32, b128 vector loads); on a hit (tag in 1..NMAX)
// gather the D=64 embedding column x[b,:,s] into e[b, tag-1, k, :].
// Bandwidth-bound part: ~36 MB streamed once.
// ---------------------------------------------------------------------------
__global__ void ae_gather(const float* __restrict__ x,
                          const int*  __restrict__ tags,
                          float*      __restrict__ e) {
  const int nvec = (NB * NK * NS) / 4;
  int v = blockIdx.x * blockDim.x + threadIdx.x;
  if (v >= nvec) return;

  const int4 t4 = ((const int4*)tags)[v];       // global_load_b128
  const int base = v * 4;
  // 4 consecutive flat indices share the same (b,k) since NS is a mult. of 4
  const int bk = base >> 16;                    // = b*NK + k
  const int b  = bk / NK;
  const int k  = bk - b * NK;
  const int s0 = base & (NS - 1);

  int tv[4] = {t4.x, t4.y, t4.z, t4.w};
#pragma unroll
  for (int j = 0; j < 4; ++j) {
    int t = tv[j];
    if (t > 0 && t <= NMAX) {
      const int s = s0 + j;
      const float* xp = x + (size_t)b * ND * NS + s;
      float* ep = e + ((((size_t)b * NMAX + (t - 1)) * NK + k) << 6);
#pragma unroll 8
      for (int d = 0; d < ND; ++d) ep[d] = xp[(size_t)d * NS];
    }
  }
}

// ---------------------------------------------------------------------------
// Kernel 2: one wave32 per batch image.
//   - mean[n][d] over K=17, pull = sum (e-mean)^2 (masked n < numH)
//   - Gram G = M * M^T via v_wmma_f32_16x16x4_f32. The mean matrix is stored
//     padded to 16 rows (rows 8..15 zeroed) so fragment loads are
//     unconditional aligned ds_load_b64 — no EXEC predication around LDS ops.
//     A-frag == B-frag because the operand is symmetric under the 16x4 / 4x16
//     VGPR layouts (lane L -> row/col L&15, K-pair selected by lane half).
//   - push = sum over ordered pairs exp(-(G[m][m]+G[n][n]-2G[m][n])/sigma^2)
// ---------------------------------------------------------------------------
__global__ __launch_bounds__(32) void ae_reduce(const float* __restrict__ e,
                                                const int*   __restrict__ numh,
                                                float*       __restrict__ out) {
  const int b    = blockIdx.x;
  const int lane = threadIdx.x;        // 0..31
  const int nh   = numh[b];

  __shared__ float meanlds[NPAD * ND]; // rows 8..15 zero-padded
  __shared__ float diag[NMAX];

  const float* eb = e + (size_t)b * NMAX * NK * ND;

  // zero the padding rows (8..15)
  for (int p = lane; p < (NPAD - NMAX) * ND; p += 32)
    meanlds[NMAX * ND + p] = 0.0f;

  float pull = 0.0f;
  for (int p = lane; p < NMAX * ND; p += 32) {   // 16 iters, no divergence
    const int n = p >> 6;
    const int d = p & 63;
    const float* ep = eb + n * NK * ND + d;
    float sum = 0.0f;
#pragma unroll
    for (int kk = 0; kk < NK; ++kk) sum += ep[kk * ND];
    const float m = sum * (1.0f / NK);
    meanlds[p] = m;
    if (n < nh) {
      float acc = 0.0f;
#pragma unroll
      for (int kk = 0; kk < NK; ++kk) { float df = ep[kk * ND] - m; acc += df * df; }
      pull += acc;
    }
  }
  __syncthreads();

  // ---- Gram matrix with f32 WMMA: 16x16 result, K=64 in 16 chunks of 4 ----
  const int hi = lane >> 4;            // selects K-pair {0,1} vs {2,3}
  const int nn = lane & 15;            // row of A / col of B held by this lane
  const float* frow = &meanlds[nn * ND + hi * 2];
  v8f acc = {};
#pragma unroll
  for (int kc = 0; kc < 16; ++kc) {
    // unconditional aligned 8-byte LDS load: ds_load_b64
    const v2f f = *(const v2f*)(frow + kc * 4);
    // D = A * B + C ; A-frag == B-frag for the symmetric Gram product
    acc = __builtin_amdgcn_wmma_f32_16x16x4_f32(
        /*neg_a=*/false, f, /*neg_b=*/false, f,
        /*c_mod=*/(short)0, acc, /*reuse_a=*/false, /*reuse_b=*/false);
  }

  // Diagonal ||mean_n||^2: held by lanes 0..7 (hi==0), register j == nn.
#pragma unroll
  for (int j = 0; j < NMAX; ++j)
    if (hi == 0 && nn == j) diag[j] = acc[j];
  __syncthreads();

  // Push: lane nn (hi==0, nn<NMAX) holds column G[*][nn] in acc[0..7].
  float push = 0.0f;
  if (hi == 0 && nn < NMAX && nn < nh) {
#pragma unroll
    for (int j = 0; j < NMAX; ++j) {
      if (j != nn && j < nh) {
        const float d2 = diag[j] + diag[nn] - 2.0f * acc[j];
        push += expf(-d2 * SIG2INV);
      }
    }
  }

  float tot = pull + push;
#pragma unroll
  for (int o = 16; o > 0; o >>= 1) tot += __shfl_down(tot, o, 32);
  if (lane == 0) atomicAdd(out, tot);
}

// ---------------------------------------------------------------------------
extern "C" void kernel_launch(void* const* d_in, const int* in_sizes, int n_in,
                              void* d_out, int out_size, void* d_ws, size_t ws_size,
                              hipStream_t stream) {
  const float* x    = (const float*)d_in[0];   // [B, D, H, W] fp32
  const int*   tags = (const int*)  d_in[1];   // [B, K, H, W] int32
  const int*   numh = (const int*)  d_in[2];   // [B, 1] int32
  float* out = (float*)d_out;                  // scalar
  float* e   = (float*)d_ws;                   // [B, NMAX, K, D] fp32 scratch

  const int etot = NB * NMAX * NK * ND;
  ae_zero<<<(etot + 255) / 256, 256, 0, stream>>>(e, out);

  const int nvec = (NB * NK * NS) / 4;
  ae_gather<<<(nvec + 255) / 256, 256, 0, stream>>>(x, tags, e);

  ae_reduce<<<NB, 32, 0, stream>>>(e, numh, out);
}